// Model_38792144617695
// MI455X (gfx1250) — compile-verified
//
#include <hip/hip_runtime.h>
#include <hip/hip_bf16.h>

#define HDIM 32
#define VVOC 16

typedef __attribute__((ext_vector_type(2))) float v2f;
typedef __attribute__((ext_vector_type(8))) float v8f;

// Broadcast lane k's value of v to all lanes as a wave-uniform (SGPR) float.
__device__ __forceinline__ float bcast(float v, int k) {
    return __int_as_float(__builtin_amdgcn_readlane(__float_as_int(v), k));
}

#if __has_builtin(__builtin_amdgcn_tanhf)
__device__ __forceinline__ float fast_tanh(float x) { return __builtin_amdgcn_tanhf(x); }
#else
__device__ __forceinline__ float fast_tanh(float x) {
    // tanh(x) = 1 - 2/(exp(2x)+1); saturates correctly at +-inf.
    float e = __expf(2.0f * x);
    return 1.0f - 2.0f * __builtin_amdgcn_rcpf(e + 1.0f);
}
#endif

__device__ __forceinline__ float fast_sigmoid(float x) {
    // sigmoid(x) = 0.5*tanh(x/2) + 0.5
    return fmaf(0.5f, fast_tanh(0.5f * x), 0.5f);
}

__global__ __launch_bounds__(32, 1)
void lstm_nalu_kernel(const int*   __restrict__ x,
                      const float* __restrict__ emb,
                      const float* __restrict__ W_ih,
                      const float* __restrict__ W_hh,
                      const float* __restrict__ b_ih,
                      const float* __restrict__ b_hh,
                      const float* __restrict__ G,
                      const float* __restrict__ W_hat,
                      const float* __restrict__ M_hat,
                      float* __restrict__ out,
                      int T)
{
    __shared__ float  raw[VVOC * 4 * HDIM];   // [16 tok][128 gate]
    __shared__ float4 tbl[VVOC * HDIM];       // [16 tok][32 lane] = (i,f,g,o) bias

    const int lane = threadIdx.x;             // 0..31, one full wave32
    const int half = lane >> 4;               // 0 or 1
    const int l16  = lane & 15;

    // ---------------- Phase A: token-bias table -------------------------
    // table[tok][gate] = emb[tok,:] . W_ih[gate,:] + b_ih[gate] + b_hh[gate]
    // Computed as fp32 WMMA: D[16 tok x 16 gate] += A[16x4] * B[4x16], 8 K-chunks,
    // 8 gate tiles -> 64x v_wmma_f32_16x16x4_f32.
#if __has_builtin(__builtin_amdgcn_wmma_f32_16x16x4_f32)
    #pragma unroll 1
    for (int nt = 0; nt < 8; ++nt) {
        v8f acc = {0.f, 0.f, 0.f, 0.f, 0.f, 0.f, 0.f, 0.f};
        const int gate = nt * 16 + l16;
        #pragma unroll
        for (int kc = 0; kc < 8; ++kc) {
            const int k0 = kc * 4 + half * 2;   // A VGPR0:K={0,2}, VGPR1:K={1,3} per half
            v2f a, b;
            a.x = emb[l16 * HDIM + k0];         // A[m=l16][k0]
            a.y = emb[l16 * HDIM + k0 + 1];
            b.x = W_ih[gate * HDIM + k0];       // B[k][n] = W_ih^T -> W_ih[gate][k]
            b.y = W_ih[gate * HDIM + k0 + 1];
            acc = __builtin_amdgcn_wmma_f32_16x16x4_f32(
                false, a, false, b, (short)0, acc, false, false);
        }
        const float bias = b_ih[gate] + b_hh[gate];
        #pragma unroll
        for (int r = 0; r < 8; ++r) {
            const int tok = half * 8 + r;       // D: lanes0-15 M=r, lanes16-31 M=8+r
            raw[tok * 128 + gate] = acc[r] + bias;
        }
    }
#else
    // VALU fallback (keeps compile alive if the f32 WMMA builtin is absent).
    #pragma unroll 1
    for (int g4 = 0; g4 < 4; ++g4) {
        const int gate = g4 * 32 + lane;
        const float bias = b_ih[gate] + b_hh[gate];
        for (int tok = 0; tok < VVOC; ++tok) {
            float s = 0.f;
            #pragma unroll
            for (int k = 0; k < HDIM; ++k)
                s = fmaf(emb[tok * HDIM + k], W_ih[gate * HDIM + k], s);
            raw[tok * 128 + gate] = s + bias;
        }
    }
#endif
    __syncthreads();
    #pragma unroll
    for (int tok = 0; tok < VVOC; ++tok) {
        float4 v;
        v.x = raw[tok * 128 + 0 * HDIM + lane];   // i
        v.y = raw[tok * 128 + 1 * HDIM + lane];   // f
        v.z = raw[tok * 128 + 2 * HDIM + lane];   // g
        v.w = raw[tok * 128 + 3 * HDIM + lane];   // o
        tbl[tok * HDIM + lane] = v;
    }
    __syncthreads();

    // ---------------- Phase B: W_hh resident in VGPRs -------------------
    // Lane l owns gate rows l (i), 32+l (f), 64+l (g), 96+l (o): 128 floats/lane.
    float wi[HDIM], wf[HDIM], wg[HDIM], wo[HDIM];
    {
        const float4* Pi = (const float4*)(W_hh + ( 0 + lane) * HDIM);
        const float4* Pf = (const float4*)(W_hh + (32 + lane) * HDIM);
        const float4* Pg = (const float4*)(W_hh + (64 + lane) * HDIM);
        const float4* Po = (const float4*)(W_hh + (96 + lane) * HDIM);
        #pragma unroll
        for (int j = 0; j < 8; ++j) {
            float4 a = Pi[j]; wi[4*j]=a.x; wi[4*j+1]=a.y; wi[4*j+2]=a.z; wi[4*j+3]=a.w;
            float4 b = Pf[j]; wf[4*j]=b.x; wf[4*j+1]=b.y; wf[4*j+2]=b.z; wf[4*j+3]=b.w;
            float4 c = Pg[j]; wg[4*j]=c.x; wg[4*j+1]=c.y; wg[4*j+2]=c.z; wg[4*j+3]=c.w;
            float4 d = Po[j]; wo[4*j]=d.x; wo[4*j+1]=d.y; wo[4*j+2]=d.z; wo[4*j+3]=d.w;
        }
    }

    // ---------------- Phase C: sequential LSTM --------------------------
    float h = 0.f, c = 0.f;
    float4 bias = tbl[x[0] * HDIM + lane];        // token biases for t=0
    for (int t = 0; t < T; ++t) {
        // Software pipeline: fetch next step's token bias now (scalar load +
        // ds_load_b128); it is consumed only after next step's matvec.
        const int tn      = (t + 1 < T) ? (t + 1) : (T - 1);
        const int ntok    = x[tn];                // uniform -> s_load
        const float4 nbias = tbl[ntok * HDIM + lane];

        float gi = 0.f, gf = 0.f, gg = 0.f, go = 0.f;
        #pragma unroll
        for (int k = 0; k < HDIM; ++k) {
            const float hk = bcast(h, k);         // v_readlane -> SGPR
            gi = fmaf(wi[k], hk, gi);
            gf = fmaf(wf[k], hk, gf);
            gg = fmaf(wg[k], hk, gg);
            go = fmaf(wo[k], hk, go);
        }
        gi += bias.x; gf += bias.y; gg += bias.z; go += bias.w;

        const float si = fast_sigmoid(gi);
        const float sf = fast_sigmoid(gf);
        const float so = fast_sigmoid(go);
        const float tg = fast_tanh(gg);
        c = fmaf(sf, c, si * tg);
        h = so * fast_tanh(c);
        bias = nbias;
    }

    // ---------------- Phase D: NALU head --------------------------------
    const float Wn = fast_tanh(W_hat[lane]) * fast_sigmoid(M_hat[lane]);
    const float pa = h * Wn;
    const float pm = __logf(fabsf(h) + 1e-7f) * Wn;
    const float pg = h * G[lane];
    float a = 0.f, m = 0.f, g = 0.f;
    #pragma unroll
    for (int k = 0; k < 32; ++k) {
        a += bcast(pa, k);
        m += bcast(pm, k);
        g += bcast(pg, k);
    }
    m = __expf(m);
    g = fast_sigmoid(g);
    if (lane == 0) out[0] = fmaf(g, a, (1.0f - g) * m);
}

extern "C" void kernel_launch(void* const* d_in, const int* in_sizes, int n_in,
                              void* d_out, int out_size, void* d_ws, size_t ws_size,
                              hipStream_t stream) {
    (void)n_in; (void)out_size; (void)d_ws; (void)ws_size;
    const int*   x     = (const int*)  d_in[0];
    const float* emb   = (const float*)d_in[1];
    const float* W_ih  = (const float*)d_in[2];
    const float* W_hh  = (const float*)d_in[3];
    const float* b_ih  = (const float*)d_in[4];
    const float* b_hh  = (const float*)d_in[5];
    const float* G     = (const float*)d_in[6];
    const float* W_hat = (const float*)d_in[7];
    const float* M_hat = (const float*)d_in[8];
    float* out = (float*)d_out;
    const int T = in_sizes[0];   // 524288 (B == 1)

    hipLaunchKernelGGL(lstm_nalu_kernel, dim3(1), dim3(32), 0, stream,
                       x, emb, W_ih, W_hh, b_ih, b_hh, G, W_hat, M_hat, out, T);
}